// GraphConvolution_Sage_1614907703383
// MI455X (gfx1250) — compile-verified
//
#include <hip/hip_runtime.h>

typedef float v2f __attribute__((ext_vector_type(2)));
typedef float v8f __attribute__((ext_vector_type(8)));

constexpr int N_NODES = 50000;
constexpr int N_EDGES = 600000;
constexpr int DIM     = 128;
constexpr int N_GRAPH = 64;
constexpr int STR     = 132;   // LDS row stride (floats): stride%64==4 -> conflict-free col reads

// ---------------- utility kernels ----------------

__global__ void zero_f4_kernel(float4* p, int n4) {
    int i = blockIdx.x * 256 + threadIdx.x;
    if (i < n4) p[i] = make_float4(0.f, 0.f, 0.f, 0.f);
}

__global__ void zero_f_kernel(float* p, int n) {
    int i = blockIdx.x * 256 + threadIdx.x;
    if (i < n) p[i] = 0.f;
}

__global__ void count_kernel(const int* __restrict__ dst, float* __restrict__ cnt, int e) {
    int i = blockIdx.x * 256 + threadIdx.x;
    if (i < e) atomicAdd(&cnt[dst[i]], 1.0f);
}

__global__ void inv_kernel(const float* __restrict__ cnt, float* __restrict__ inv, int n) {
    int i = blockIdx.x * 256 + threadIdx.x;
    if (i < n) inv[i] = 1.0f / fmaxf(cnt[i], 1.0f);
}

// ---------------- edge scatter: agg[dst] += x[src] ----------------
// 32 lanes per edge, each lane owns a float4 chunk of the 128-wide row.
__global__ void scatter_kernel(const float* __restrict__ x,
                               const int* __restrict__ src,
                               const int* __restrict__ dst,
                               float* __restrict__ agg, int e) {
    int t = blockIdx.x * 256 + threadIdx.x;
    int edge = t >> 5;
    int c    = (t & 31) << 2;
    if (edge < e) {
        int s = src[edge];
        int d = dst[edge];
        const float4 v = *(const float4*)&x[(size_t)s * DIM + c];
        float* p = &agg[(size_t)d * DIM + c];
        atomicAdd(p + 0, v.x);
        atomicAdd(p + 1, v.y);
        atomicAdd(p + 2, v.z);
        atomicAdd(p + 3, v.w);
    }
}

// ---------------- fused SAGE layer GEMM (f32 WMMA) ----------------
// Block = 256 threads = 8 wave32s; one block per 16-node tile.
// Wave w computes output columns [16w, 16w+16).
// out = leaky( (agg*inv_cnt) @ Wl^T + bl + x @ Wr^T ) (+ x if residual)
__global__ __launch_bounds__(256) void sage_gemm_kernel(
        const float* __restrict__ xin, const float* __restrict__ agg,
        const float* __restrict__ inv_cnt,
        const float* __restrict__ Wl, const float* __restrict__ Wr,
        const float* __restrict__ bias, float* __restrict__ xout,
        int residual) {
    __shared__ float sX[16 * STR];
    __shared__ float sG[16 * STR];

    const int tid  = threadIdx.x;
    const size_t base = (size_t)blockIdx.x * 16;

    // Stage the 16x128 x-tile and normalized agg-tile into LDS.
    {
        int r  = tid >> 4;          // 0..15 (node row)
        int c0 = (tid & 15) * 8;    // 0..120
        const float4* px = (const float4*)&xin[(base + r) * DIM + c0];
        float4 v0 = px[0], v1 = px[1];
        *(float4*)&sX[r * STR + c0]     = v0;
        *(float4*)&sX[r * STR + c0 + 4] = v1;

        float s = inv_cnt[base + r];
        const float4* pg = (const float4*)&agg[(base + r) * DIM + c0];
        float4 g0 = pg[0], g1 = pg[1];
        g0.x *= s; g0.y *= s; g0.z *= s; g0.w *= s;
        g1.x *= s; g1.y *= s; g1.z *= s; g1.w *= s;
        *(float4*)&sG[r * STR + c0]     = g0;
        *(float4*)&sG[r * STR + c0 + 4] = g1;
    }
    __syncthreads();

    const int wv   = tid >> 5;           // wave id -> output col tile
    const int lane = tid & 31;
    const int ln   = lane & 15;          // row (A) / col (B) within tile
    const int khi  = (lane >> 4) << 1;   // 0 or 2: K sub-offset per ISA layout
    const int ncol = wv * 16 + ln;       // global output column

    const float* wlrow = &Wl[(size_t)ncol * DIM];  // B[k][n] = Wl[n][k]
    const float* wrrow = &Wr[(size_t)ncol * DIM];

    v8f acc = {};
#pragma unroll
    for (int k0 = 0; k0 < DIM; k0 += 4) {
        int kk = k0 + khi;
        v2f aG = *(const v2f*)&sG[ln * STR + kk];
        v2f aX = *(const v2f*)&sX[ln * STR + kk];
        v2f bL = *(const v2f*)&wlrow[kk];
        v2f bR = *(const v2f*)&wrrow[kk];
        acc = __builtin_amdgcn_wmma_f32_16x16x4_f32(false, aG, false, bL,
                                                    (short)0, acc, false, false);
        acc = __builtin_amdgcn_wmma_f32_16x16x4_f32(false, aX, false, bR,
                                                    (short)0, acc, false, false);
    }

    // Epilogue: C/D layout -> VGPR v holds row v (lanes 0-15) / row v+8 (lanes 16-31).
    const float bcol  = bias[ncol];
    const int   rbase = (lane >> 4) * 8;
#pragma unroll
    for (int v = 0; v < 8; ++v) {
        int row = rbase + v;
        float val = acc[v] + bcol;
        val = val >= 0.f ? val : 0.01f * val;   // leaky_relu
        if (residual) val += sX[row * STR + ncol];
        xout[(base + row) * DIM + ncol] = val;
    }
}

// ---------------- pooling (batch is sorted) ----------------
__global__ void pool_accum_kernel(const float* __restrict__ x,
                                  const int* __restrict__ batch,
                                  float* __restrict__ pool,
                                  float* __restrict__ cntg, int n) {
    int c  = threadIdx.x;           // 0..127 (column)
    int n0 = blockIdx.x * 128;
    int n1 = min(n0 + 128, n);
    int cur = batch[n0];
    float acc = 0.f, cnt = 0.f;
    for (int i = n0; i < n1; ++i) {
        int g = batch[i];
        if (g != cur) {
            atomicAdd(&pool[cur * DIM + c], acc);
            if (c == 0) atomicAdd(&cntg[cur], cnt);
            acc = 0.f; cnt = 0.f; cur = g;
        }
        acc += x[(size_t)i * DIM + c];
        cnt += 1.f;
    }
    atomicAdd(&pool[cur * DIM + c], acc);
    if (c == 0) atomicAdd(&cntg[cur], cnt);
}

__global__ void pool_final_kernel(const float* __restrict__ pool,
                                  const float* __restrict__ cntg,
                                  float* __restrict__ out) {
    int i = blockIdx.x * 256 + threadIdx.x;
    if (i < N_GRAPH * DIM) {
        int g = i >> 7;
        out[i] = pool[i] / fmaxf(cntg[g], 1.0f);
    }
}

// ---------------- host launcher ----------------

extern "C" void kernel_launch(void* const* d_in, const int* in_sizes, int n_in,
                              void* d_out, int out_size, void* d_ws, size_t ws_size,
                              hipStream_t stream) {
    const float* x0    = (const float*)d_in[0];
    const int*   ei    = (const int*)d_in[1];
    const int*   batch = (const int*)d_in[2];
    const float* Wl    = (const float*)d_in[3];
    const float* bl    = (const float*)d_in[4];
    const float* Wr    = (const float*)d_in[5];
    float* out = (float*)d_out;

    const int* src = ei;
    const int* dst = ei + N_EDGES;

    char* ws = (char*)d_ws;
    const size_t XB = (size_t)N_NODES * DIM * sizeof(float);   // 25.6 MB
    float* agg  = (float*)(ws);
    float* xa   = (float*)(ws + XB);
    float* xb   = (float*)(ws + 2 * XB);
    float* inv  = (float*)(ws + 3 * XB);                        // N floats
    float* cnt  = (float*)(ws + 3 * XB + 256 * 1024);           // N floats
    float* pool = (float*)(ws + 3 * XB + 512 * 1024);           // G*D floats
    float* cntg = pool + N_GRAPH * DIM;                         // G floats (contiguous)

    const int nfloats = N_NODES * DIM;          // 6.4M
    const int nf4     = nfloats / 4;            // 1.6M

    // In-degree counts -> reciprocals (shared by all layers)
    zero_f_kernel<<<(N_NODES + 255) / 256, 256, 0, stream>>>(cnt, N_NODES);
    count_kernel<<<(N_EDGES + 255) / 256, 256, 0, stream>>>(dst, cnt, N_EDGES);
    inv_kernel<<<(N_NODES + 255) / 256, 256, 0, stream>>>(cnt, inv, N_NODES);

    const float* xin = x0;
    float* bufs[2] = { xa, xb };
    const int scat_blocks = (N_EDGES * 32 + 255) / 256;

    for (int layer = 0; layer < 4; ++layer) {
        zero_f4_kernel<<<(nf4 + 255) / 256, 256, 0, stream>>>((float4*)agg, nf4);
        scatter_kernel<<<scat_blocks, 256, 0, stream>>>(xin, src, dst, agg, N_EDGES);
        sage_gemm_kernel<<<N_NODES / 16, 256, 0, stream>>>(
            xin, agg, inv,
            Wl + (size_t)layer * DIM * DIM,
            Wr + (size_t)layer * DIM * DIM,
            bl + (size_t)layer * DIM,
            bufs[layer & 1], layer >= 2 ? 1 : 0);
        xin = bufs[layer & 1];
    }

    // Global mean pool
    zero_f_kernel<<<(N_GRAPH * DIM + N_GRAPH + 255) / 256, 256, 0, stream>>>(
        pool, N_GRAPH * DIM + N_GRAPH);
    pool_accum_kernel<<<(N_NODES + 127) / 128, 128, 0, stream>>>(xin, batch, pool, cntg, N_NODES);
    pool_final_kernel<<<(N_GRAPH * DIM + 255) / 256, 256, 0, stream>>>(pool, cntg, out);
}